// Prop_72292889526826
// MI455X (gfx1250) — compile-verified
//
#include <hip/hip_runtime.h>
#include <math.h>

// ---------------- problem constants (from reference) ----------------
#define NYV   600
#define NXV   250
#define PADW  10
#define NYP   (NYV + 2*PADW)     // 620
#define NXP   (NXV + 2*PADW)     // 270
#define NSHOT 4
#define NSRC  1
#define NREC  100
#define NTT   300
#define DXX   4.0f
#define DTT   0.004f
#define MINV  1000.0f
#define MAXV  2500.0f
#define INV_DX2 (1.0f/(DXX*DXX))
#define FIRLEN 33
#define TAPERN 100
#define SRROWS (NSHOT*NREC)        // 400 rows (= 25 * 16)
#define XPC    (NTT + FIRLEN - 1)  // 332 padded trace length
#define WNC    304                 // 300 cols padded to 19*16
#define FLD    (NSHOT*NYP*NXP)     // 669600 cells
#define PI_F   3.14159265358979323846f

typedef float v2f __attribute__((ext_vector_type(2)));
typedef float v8f __attribute__((ext_vector_type(8)));

// ---------------- init: v2dt2, PML coefficient vectors, zero fields ----------------
__device__ __forceinline__ float pml_sigma(int i, int n) {
    float smax = 3.0f * MAXV * logf(1000.0f) / (2.0f * PADW * DXX);
    float d = fmaxf((float)(PADW - i), (float)(i - (n - 1 - PADW))) / (float)PADW;
    d = fminf(fmaxf(d, 0.0f), 1.0f);
    return smax * d * d;
}

__global__ void init_kernel(const float* __restrict__ vlogit,
                            float* __restrict__ f0x, float* __restrict__ f0y,
                            float* __restrict__ f1x, float* __restrict__ f1y,
                            float* __restrict__ u0,  float* __restrict__ u1,
                            float* __restrict__ v2dt2,
                            float* __restrict__ axm, float* __restrict__ axpi,
                            float* __restrict__ aym, float* __restrict__ aypi) {
    int gid = blockIdx.x * blockDim.x + threadIdx.x;
    if (gid < FLD) {
        f0x[gid] = 0.f; f0y[gid] = 0.f; f1x[gid] = 0.f; f1y[gid] = 0.f;
        u0[gid] = 0.f;  u1[gid] = 0.f;
    }
    if (gid < NYP * NXP) {
        int x = gid % NXP, y = gid / NXP;
        int yy = min(max(y - PADW, 0), NYV - 1);
        int xx = min(max(x - PADW, 0), NXV - 1);
        float s = 1.0f / (1.0f + expf(-vlogit[yy * NXV + xx]));
        float v = s * (MAXV - MINV) + MINV;
        float vd = v * DTT;
        v2dt2[gid] = vd * vd;
    }
    if (gid < NXP) {
        float sx = pml_sigma(gid, NXP);
        axm[gid]  = 1.0f - 0.5f * DTT * sx;
        axpi[gid] = 1.0f / (1.0f + 0.5f * DTT * sx);
    }
    if (gid < NYP) {
        float sy = pml_sigma(gid, NYP);
        aym[gid]  = 1.0f - 0.5f * DTT * sy;
        aypi[gid] = 1.0f / (1.0f + 0.5f * DTT * sy);
    }
}

// ---------------- one leapfrog step ----------------
// ucur = ux+uy of the current (newest) fields; stencil reads only ucur.
// Writes new split fields in place over prev (only read at own cell) and
// writes unew = uxn+uyn (post-injection) for the next step / recorder.
__global__ void step_kernel(const float* __restrict__ ucur, float* __restrict__ unew,
                            const float* __restrict__ cux, const float* __restrict__ cuy,
                            float* __restrict__ pux, float* __restrict__ puy,
                            const float* __restrict__ v2dt2,
                            const float* __restrict__ axm, const float* __restrict__ axpi,
                            const float* __restrict__ aym, const float* __restrict__ aypi,
                            const float* __restrict__ amps, const int* __restrict__ sloc,
                            const int* __restrict__ rloc, float* __restrict__ recs, int t) {
    int gid = blockIdx.x * blockDim.x + threadIdx.x;
    // record previous step's wavefield at receivers (reads this kernel's RO input ucur)
    if (gid < SRROWS && t > 0) {
        int ry = rloc[gid * 2 + 0] + PADW;
        int rx = rloc[gid * 2 + 1] + PADW;
        int s  = gid / NREC;
        int fi = (s * NYP + ry) * NXP + rx;
        recs[gid * NTT + (t - 1)] = ucur[fi];
    }
    if (gid >= FLD) return;
    int x  = gid % NXP;
    int yy = gid / NXP;
    int y  = yy % NYP;
    int s  = yy / NYP;

    float ux = cux[gid], uy = cuy[gid];
    float u = ux + uy;
    float d2x = 0.0f, d2y = 0.0f;
    if (x > 0 && x < NXP - 1)
        d2x = (ucur[gid + 1] - 2.0f * u + ucur[gid - 1]) * INV_DX2;
    if (y > 0 && y < NYP - 1)
        d2y = (ucur[gid + NXP] - 2.0f * u + ucur[gid - NXP]) * INV_DX2;

    float c2  = v2dt2[y * NXP + x];
    float uxn = (2.0f * ux - axm[x] * pux[gid] + c2 * d2x) * axpi[x];
    float uyn = (2.0f * uy - aym[y] * puy[gid] + c2 * d2y) * aypi[y];

    int sy = sloc[s * 2 + 0] + PADW;
    int sx = sloc[s * 2 + 1] + PADW;
    if (y == sy && x == sx) {
        float inj = 0.5f * c2 * amps[s * NTT + t];
        uxn += inj; uyn += inj;
    }
    pux[gid] = uxn;
    puy[gid] = uyn;
    unew[gid] = uxn + uyn;
}

// ---------------- final receiver record for t = NTT-1 ----------------
__global__ void final_record_kernel(const float* __restrict__ ucur,
                                    const int* __restrict__ rloc, float* __restrict__ recs) {
    int gid = blockIdx.x * blockDim.x + threadIdx.x;
    if (gid >= SRROWS) return;
    int ry = rloc[gid * 2 + 0] + PADW;
    int rx = rloc[gid * 2 + 1] + PADW;
    int s  = gid / NREC;
    int fi = (s * NYP + ry) * NXP + rx;
    recs[gid * NTT + (NTT - 1)] = ucur[fi];
}

// ---------------- FIR taps (windowed sinc + Hamming, normalized) ----------------
__global__ void taps_kernel(float* __restrict__ taps) {
    __shared__ float h[FIRLEN];
    __shared__ float ssum;
    int k = threadIdx.x;
    if (k < FIRLEN) {
        float fc = 30.0f * DTT;                 // normalized cutoff = 0.12
        float n  = (float)k - (FIRLEN - 1) * 0.5f;
        float x  = 2.0f * fc * n;
        float sc = (fabsf(x) < 1e-8f) ? 1.0f : sinf(PI_F * x) / (PI_F * x);
        float w  = 0.54f - 0.46f * cosf(2.0f * PI_F * (float)k / (float)(FIRLEN - 1));
        h[k] = 2.0f * fc * sc * w;
    }
    __syncthreads();
    if (k == 0) { float t = 0.f; for (int i = 0; i < FIRLEN; ++i) t += h[i]; ssum = t; }
    __syncthreads();
    if (k < FIRLEN) taps[k] = h[k] / ssum;
}

// ---------------- banded FIR matrix W[XPC][WNC], W[i][t] = h[i-t] ----------------
__global__ void wfill_kernel(const float* __restrict__ taps, float* __restrict__ W) {
    int gid = blockIdx.x * blockDim.x + threadIdx.x;
    if (gid >= XPC * WNC) return;
    int t = gid % WNC, i = gid / WNC;
    int d = i - t;
    float v = 0.0f;
    if (t < NTT && d >= 0 && d < FIRLEN) v = taps[d];
    W[gid] = v;
}

// ---------------- padded + tapered trace matrix Xp[400][332] ----------------
__global__ void xp_kernel(const float* __restrict__ recs, float* __restrict__ Xp) {
    int gid = blockIdx.x * blockDim.x + threadIdx.x;
    if (gid >= SRROWS * XPC) return;
    int j = gid % XPC, sr = gid / XPC;
    int t = j - (FIRLEN - 1) / 2;
    float v = 0.0f;
    if (t >= 0 && t < NTT) {
        float w = 1.0f;
        if (t >= NTT - TAPERN)
            w = 0.5f * (1.0f + cosf(PI_F * (float)(t - (NTT - TAPERN) + 1) / (float)TAPERN));
        v = recs[sr * NTT + t] * w;
    }
    Xp[gid] = v;
}

// ---------------- FIR as banded GEMM on f32 WMMA (16x16x4) ----------------
// One wave per 16x16 output tile (EXEC all-ones around WMMA per ISA restriction).
// Fragment layout per CDNA5 ISA tables: vreg r of a K=4 fragment holds rows
// K=r (lanes 0-15) and K=r+2 (lanes 16-31); D vreg r holds M=r / M=r+8.
__global__ void __launch_bounds__(32)
fir_wmma_kernel(const float* __restrict__ Xp, const float* __restrict__ W,
                float* __restrict__ out) {
    const int NTI = WNC / 16;      // 19
    int tile = blockIdx.x;
    int mt = tile / NTI, ntile = tile % NTI;
    int m0 = mt * 16, n0 = ntile * 16;
    int lane = threadIdx.x & 31;
    int half = lane >> 4;          // 0: K/M low half, 1: high half
    int l    = lane & 15;

    v8f acc = {};
    int kend = n0 + FIRLEN + 15;   // band: rows [n0, n0+47]
    if (kend > XPC) kend = XPC;    // span stays divisible by 4
    for (int k = n0; k < kend; k += 4) {
        int k0 = k + half * 2;     // K index held in vreg 0 for this lane
        int k1 = k0 + 1;           // K index held in vreg 1
        v2f a, b;
        a.x = Xp[(m0 + l) * XPC + k0];
        a.y = Xp[(m0 + l) * XPC + k1];
        b.x = W[k0 * WNC + (n0 + l)];
        b.y = W[k1 * WNC + (n0 + l)];
        acc = __builtin_amdgcn_wmma_f32_16x16x4_f32(
            /*neg_a=*/false, a, /*neg_b=*/false, b,
            /*c_mod=*/(short)0, acc, /*reuse_a=*/false, /*reuse_b=*/false);
    }
    int col = n0 + l;
    if (col < NTT) {
#pragma unroll
        for (int r = 0; r < 8; ++r) {
            int row = m0 + r + half * 8;
            out[row * NTT + col] = acc[r];
        }
    }
}

// ---------------- host orchestration ----------------
extern "C" void kernel_launch(void* const* d_in, const int* in_sizes, int n_in,
                              void* d_out, int out_size, void* d_ws, size_t ws_size,
                              hipStream_t stream) {
    const float* v_logit = (const float*)d_in[0];
    const float* amps    = (const float*)d_in[1];   // [4,1,300]
    const int*   sloc    = (const int*)d_in[2];     // [4,1,2]
    const int*   rloc    = (const int*)d_in[3];     // [4,100,2]
    float* out = (float*)d_out;                     // [4,100,300]

    float* ws = (float*)d_ws;
    size_t o = 0;
    float* f0x   = ws + o; o += FLD;
    float* f0y   = ws + o; o += FLD;
    float* f1x   = ws + o; o += FLD;
    float* f1y   = ws + o; o += FLD;
    float* u0    = ws + o; o += FLD;
    float* u1    = ws + o; o += FLD;
    float* v2    = ws + o; o += (size_t)NYP * NXP;
    float* axm   = ws + o; o += NXP;
    float* axpi  = ws + o; o += NXP;
    float* aym   = ws + o; o += NYP;
    float* aypi  = ws + o; o += NYP;
    float* recs  = ws + o; o += (size_t)SRROWS * NTT;
    float* xp    = ws + o; o += (size_t)SRROWS * XPC;
    float* wmat  = ws + o; o += (size_t)XPC * WNC;
    float* taps  = ws + o; o += 64;

    int blocks = (FLD + 255) / 256;
    init_kernel<<<blocks, 256, 0, stream>>>(v_logit, f0x, f0y, f1x, f1y, u0, u1,
                                            v2, axm, axpi, aym, aypi);

    float* cx = f0x; float* cy = f0y;   // current (newest) split fields
    float* px = f1x; float* py = f1y;   // previous fields (overwritten -> new)
    float* uc = u0;  float* un = u1;    // current / next combined wavefield
    for (int t = 0; t < NTT; ++t) {
        step_kernel<<<blocks, 256, 0, stream>>>(uc, un, cx, cy, px, py, v2,
                                                axm, axpi, aym, aypi,
                                                amps, sloc, rloc, recs, t);
        float* tmp;
        tmp = cx; cx = px; px = tmp;
        tmp = cy; cy = py; py = tmp;
        tmp = uc; uc = un; un = tmp;
    }
    final_record_kernel<<<(SRROWS + 255) / 256, 256, 0, stream>>>(uc, rloc, recs);

    taps_kernel<<<1, 64, 0, stream>>>(taps);
    wfill_kernel<<<(XPC * WNC + 255) / 256, 256, 0, stream>>>(taps, wmat);
    xp_kernel<<<(SRROWS * XPC + 255) / 256, 256, 0, stream>>>(recs, xp);

    fir_wmma_kernel<<<(SRROWS / 16) * (WNC / 16), 32, 0, stream>>>(xp, wmat, out);
}